// ConvTranspose3dMPS_43258910605943
// MI455X (gfx1250) — compile-verified
//
#include <hip/hip_runtime.h>
#include <hip/hip_bf16.h>

typedef __attribute__((ext_vector_type(2))) float v2f;
typedef __attribute__((ext_vector_type(8))) float v8f;

// ---------------------------------------------------------------------------
// Kernel 1: collapse 3x3x3 taps onto the 2x2x2 distinct upsampled-source
// offsets, per output parity.  wc layout: [p][o][co][ci], p,o in {0..7}
// (bit2=z, bit1=y, bit0=x), co 0..31, ci 0..63.  8*8*32*64 f32 = 512 KB.
// ---------------------------------------------------------------------------
__global__ __launch_bounds__(256) void build_wc_kernel(
    const float* __restrict__ w, float* __restrict__ wc) {
  int idx = blockIdx.x * 256 + threadIdx.x;      // 131072 total
  int ci = idx & 63;
  int co = (idx >> 6) & 31;
  int o  = (idx >> 11) & 7;
  int p  = (idx >> 14) & 7;

  int pz = (p >> 2) & 1, py = (p >> 1) & 1, px = p & 1;
  int oz = (o >> 2) & 1, oy = (o >> 1) & 1, ox = o & 1;

  // per-dim contributing tap ranges: (parity,offset) -> [start, start+count)
  const int s[2][2] = {{0, 1}, {0, 2}};
  const int c[2][2] = {{1, 2}, {2, 1}};

  const float* wb = w + ((size_t)co * 64 + ci) * 27;  // [co][ci][3][3][3]
  float acc = 0.f;
  for (int kd = s[pz][oz]; kd < s[pz][oz] + c[pz][oz]; ++kd)
    for (int kh = s[py][oy]; kh < s[py][oy] + c[py][oy]; ++kh)
      for (int kw = s[px][ox]; kw < s[px][ox] + c[px][ox]; ++kw)
        acc += wb[kd * 9 + kh * 3 + kw];
  wc[idx] = acc;
}

// ---------------------------------------------------------------------------
// Kernel 0b: transpose x [b][ci][z][y][x] -> xT [b][z][y][x][ci] so the
// WMMA B-matrix K-pair (adjacent ci) becomes one aligned b64 load.
// 1M f32 = 4 MB; one-time ~8 MB of L2/HBM traffic.
// ---------------------------------------------------------------------------
__global__ __launch_bounds__(256) void transpose_x_kernel(
    const float* __restrict__ x, float* __restrict__ xT) {
  int idx = blockIdx.x * 256 + threadIdx.x;  // 1048576 total
  int ci  = idx & 63;
  int pos = idx >> 6;        // b*4096 + spatial
  int b   = pos >> 12;
  int sp  = pos & 4095;
  xT[idx] = x[((size_t)(b * 64 + ci)) * 4096 + sp];
}

// ---------------------------------------------------------------------------
// Kernel 2: per (batch, parity, tz) block; 8 waves, each owning 2 ty rows.
// Wave computes D[32co x 16tx] via V_WMMA_F32_16X16X4_F32, K = 64ci x 8off.
//   A (16x4 f32): lane m=lane&15 supplies wc[co=m (+16)], k = 2*(lane>>4)+{0,1}
//   B (4x16 f32): lane n=lane&15 supplies x[ci0+k] at wrapped position
//   D (16x16 f32): acc[r] -> co = r + 8*(lane>=16), n = lane&15
// XT=true: x arg points at ci-innermost transposed copy (single b64 B feed).
// ---------------------------------------------------------------------------
template <bool XT>
__global__ __launch_bounds__(256) void convt3d_wmma_f32_kernel(
    const float* __restrict__ x,    // XT? [4][16][16][16][64] : [4][64][16^3]
    const float* __restrict__ wc,   // [8][8][32][64]
    float* __restrict__ out) {      // [4][32][32][32][32]
  int blk  = blockIdx.x;            // 512 blocks
  int b    = blk >> 7;
  int p    = (blk >> 4) & 7;
  int tz   = blk & 15;
  int wave = threadIdx.x >> 5;
  int lane = threadIdx.x & 31;
  int n    = lane & 15;             // N index (tx) / M index (co) for A
  int kb   = (lane >> 4) << 1;      // K pair base: 0 or 2

  int pz = (p >> 2) & 1, py = (p >> 1) & 1, px = p & 1;

  v8f acc[2][2] = {};               // [ty_sub][m_tile]

  const float* wc_p = wc + (size_t)p * 8 * 32 * 64;

  for (int ci0 = 0; ci0 < 64; ci0 += 4) {
#pragma unroll
    for (int o = 0; o < 8; ++o) {
      int oz = (o >> 2) & 1, oy = (o >> 1) & 1, ox = o & 1;

      // A tiles: contiguous K-pair in ci-innermost wc layout (b64 loads)
      const float* wcb = wc_p + (size_t)o * 32 * 64 + ci0 + kb;
      v2f a0 = *(const v2f*)(wcb + (size_t)n * 64);          // co = n
      v2f a1 = *(const v2f*)(wcb + (size_t)(n + 16) * 64);   // co = n+16

      int iz = (tz - oz) & 15;
      int ix = (n - ox) & 15;

#pragma unroll
      for (int t = 0; t < 2; ++t) {
        int ty = wave * 2 + t;
        int iy = (ty - oy) & 15;
        v2f bm;
        if constexpr (XT) {
          // single aligned b64: adjacent ci are contiguous in xT
          bm = *(const v2f*)(
              x + ((size_t)b * 4096 + iz * 256 + iy * 16 + ix) * 64 + ci0 + kb);
        } else {
          const float* xb =
              x + ((((size_t)b * 64 + ci0 + kb) * 16 + iz) * 16 + iy) * 16 + ix;
          bm.x = xb[0];             // ci = ci0+kb
          bm.y = xb[4096];          // ci = ci0+kb+1 (stride 16^3)
        }

        acc[t][0] = __builtin_amdgcn_wmma_f32_16x16x4_f32(
            false, a0, false, bm, (short)0, acc[t][0], false, false);
        acc[t][1] = __builtin_amdgcn_wmma_f32_16x16x4_f32(
            false, a1, false, bm, (short)0, acc[t][1], false, false);
      }
    }
  }

  // Scatter D tiles to the strided (parity-interleaved) output.
  int mhalf = (lane >> 4) * 8;
  int zo = 2 * tz + pz;
  int xo = 2 * n + px;
#pragma unroll
  for (int t = 0; t < 2; ++t) {
    int ty = wave * 2 + t;
    int yo = 2 * ty + py;
#pragma unroll
    for (int mt = 0; mt < 2; ++mt) {
#pragma unroll
      for (int r = 0; r < 8; ++r) {
        int co = mt * 16 + mhalf + r;
        out[(((size_t)b * 32 + co) * 32 + zo) * 1024 + (size_t)yo * 32 + xo] =
            acc[t][mt][r];
      }
    }
  }
}

// ---------------------------------------------------------------------------
// Host launcher
// ---------------------------------------------------------------------------
extern "C" void kernel_launch(void* const* d_in, const int* in_sizes, int n_in,
                              void* d_out, int out_size, void* d_ws,
                              size_t ws_size, hipStream_t stream) {
  const float* x = (const float*)d_in[0];   // [4][64][16][16][16]
  const float* w = (const float*)d_in[1];   // [32][64][3][3][3]
  float* out = (float*)d_out;               // [4][32][32][32][32]

  const size_t WC_ELEMS = 8 * 8 * 32 * 64;  // 131072 (512 KB)
  const size_t XT_ELEMS = 4 * 64 * 4096;    // 1048576 (4 MB)

  float* wc = (float*)d_ws;
  build_wc_kernel<<<(int)(WC_ELEMS / 256), 256, 0, stream>>>(w, wc);

  if (ws_size >= (WC_ELEMS + XT_ELEMS) * sizeof(float)) {
    float* xT = wc + WC_ELEMS;
    transpose_x_kernel<<<(int)(XT_ELEMS / 256), 256, 0, stream>>>(x, xT);
    convt3d_wmma_f32_kernel<true><<<512, 256, 0, stream>>>(xT, wc, out);
  } else {
    convt3d_wmma_f32_kernel<false><<<512, 256, 0, stream>>>(x, wc, out);
  }
}